// FraudQuantumKernel_65481071410699
// MI455X (gfx1250) — compile-verified
//
#include <hip/hip_runtime.h>
#include <hip/hip_bf16.h>
#include <stdint.h>

// ---------------------------------------------------------------------------
// FraudQuantumKernel for MI455X (gfx1250)
//   out[i] = exp(-|x_i - y_i|^2) * net(x_i) * net(y_i)
// Memory floor: 160MB / 23.3TB/s ~= 7us. Compute is tanh-dominated unless we
// use gfx1250's native v_tanh_f32. Data movement: per-wave TDM (tensor DMA)
// double-buffered tiles into LDS; ds_load_b128 reads; NT b64 stores.
// WMMA deliberately unused: 2-wide GEMMs waste 15/16 of a 16x16 tile.
// ---------------------------------------------------------------------------

typedef __attribute__((ext_vector_type(2))) float v2f;
typedef __attribute__((ext_vector_type(4))) float v4f;
typedef __attribute__((ext_vector_type(4))) unsigned int v4u;
typedef __attribute__((ext_vector_type(4))) int v4i;
typedef __attribute__((ext_vector_type(8))) int v8i;

#define TILE_PAIRS        512          // pairs per TDM tile
#define TILE_FLOATS       1024         // floats per array per tile
#define TILE_BYTES        4096
#define BUF_STRIDE_FLOATS 2048         // [X tile | Y tile]
#define BUF_STRIDE_BYTES  8192
#define SHMEM_BYTES       16384        // 2 buffers * (X+Y)

#if __has_builtin(__builtin_amdgcn_tensor_load_to_lds) && \
    __has_builtin(__builtin_amdgcn_s_wait_tensorcnt)
#define USE_TDM 1
#else
#define USE_TDM 0
#endif

// ---- fast transcendentals (gfx1250 hardware trans ops) --------------------
__device__ __forceinline__ float fast_tanh(float v) {
#if __has_builtin(__builtin_amdgcn_tanhf)
  return __builtin_amdgcn_tanhf(v);
#elif __has_builtin(__builtin_amdgcn_tanh_f32)
  return __builtin_amdgcn_tanh_f32(v);
#else
  float r;
  asm("v_tanh_f32 %0, %1" : "=v"(r) : "v"(v));
  return r;
#endif
}

__device__ __forceinline__ float fast_exp2(float v) {
#if __has_builtin(__builtin_amdgcn_exp2f)
  return __builtin_amdgcn_exp2f(v);
#else
  return exp2f(v);  // valid on both host parse and device (ocml)
#endif
}

// ---- TDM: 1-D tile load (global -> LDS), descriptor per CDNA5 ISA
#if USE_TDM
__device__ __forceinline__ void tdm_issue_load(unsigned ldsByte,
                                               const float* gptr,
                                               unsigned tileFloats,
                                               unsigned validFloats) {
  uint64_t ga = (uint64_t)(uintptr_t)gptr;
  // group0: count=1 (valid), lds_addr, global_addr[56:0], type=2 ("image")
  v4u g0;
  g0[0] = 1u;
  g0[1] = ldsByte;
  g0[2] = (unsigned)(ga & 0xFFFFFFFFu);
  g0[3] = (unsigned)((ga >> 32) & 0x1FFFFFFu) | (2u << 30);
  // group1: data_size=2 (4B), workgroup_mask=0 (not in cluster),
  //         tensor_dim0=validFloats (OOB -> zero fill), tile_dim0=tileFloats,
  //         tile_dim1/2 = 0 (1-D), strides unused.
  v8i g1;
  g1[0] = (int)(2u << 16);                            // data_size bits 17:16
  g1[1] = (int)((validFloats & 0xFFFFu) << 16);       // tensor_dim0[15:0] @63:48
  g1[2] = (int)((validFloats >> 16) & 0xFFFFu);       // tensor_dim0[31:16] @79:64
  g1[3] = (int)((tileFloats & 0xFFFFu) << 16);        // tile_dim0 @127:112
  g1[4] = 0; g1[5] = 0; g1[6] = 0; g1[7] = 0;
  v4i z4 = {0, 0, 0, 0};
#if defined(__clang_major__) && (__clang_major__ >= 23)
  v8i z8 = {0, 0, 0, 0, 0, 0, 0, 0};
  __builtin_amdgcn_tensor_load_to_lds(g0, g1, z4, z4, z8, 0);
#else
  __builtin_amdgcn_tensor_load_to_lds(g0, g1, z4, z4, 0);
#endif
}
#endif

template <int K>
__device__ __forceinline__ void wait_tensor() {
#if USE_TDM
  __builtin_amdgcn_s_wait_tensorcnt(K);
#endif
  asm volatile("" ::: "memory");  // keep LDS reads after the wait
}

// ---- the 9-layer 2-wide MLP + 2->1 head, fully unrolled -------------------
__device__ __forceinline__ float fraud_net(float z0, float z1,
                                           const float (&W)[9][4],
                                           const float (&B)[9][2],
                                           const float (&S)[9][2],
                                           const float (&T)[9][2],
                                           float wf0, float wf1, float bfv) {
#pragma unroll
  for (int l = 0; l < 9; ++l) {
    float h0 = fmaf(z1, W[l][1], fmaf(z0, W[l][0], B[l][0]));
    float h1 = fmaf(z1, W[l][3], fmaf(z0, W[l][2], B[l][1]));
    float t0 = fast_tanh(h0);
    float t1 = fast_tanh(h1);
    z0 = fmaf(t0, S[l][0], T[l][0]);
    z1 = fmaf(t1, S[l][1], T[l][1]);
  }
  return fmaf(z1, wf1, fmaf(z0, wf0, bfv));
}

__global__ __launch_bounds__(32) void fraud_rbf_kernel(
    const float* __restrict__ x, const float* __restrict__ y,
    const float* __restrict__ W0, const float* __restrict__ b0,
    const float* __restrict__ scale0, const float* __restrict__ shift0,
    const float* __restrict__ Ws, const float* __restrict__ bs,
    const float* __restrict__ scales, const float* __restrict__ shifts,
    const float* __restrict__ Wf, const float* __restrict__ bf,
    float* __restrict__ out, int N, int numTiles) {
#if USE_TDM
  extern __shared__ float smbase[];  // dynamic LDS starts at offset 0
#endif

  // ---- preload all 93 uniform parameters into registers (scalar loads) ----
  float Wp[9][4], Bp[9][2], Sp[9][2], Tp[9][2];
#pragma unroll
  for (int k = 0; k < 4; ++k) Wp[0][k] = W0[k];
  Bp[0][0] = b0[0];     Bp[0][1] = b0[1];
  Sp[0][0] = scale0[0]; Sp[0][1] = scale0[1];
  Tp[0][0] = shift0[0]; Tp[0][1] = shift0[1];
#pragma unroll
  for (int l = 0; l < 8; ++l) {
#pragma unroll
    for (int k = 0; k < 4; ++k) Wp[l + 1][k] = Ws[l * 4 + k];
    Bp[l + 1][0] = bs[l * 2];     Bp[l + 1][1] = bs[l * 2 + 1];
    Sp[l + 1][0] = scales[l * 2]; Sp[l + 1][1] = scales[l * 2 + 1];
    Tp[l + 1][0] = shifts[l * 2]; Tp[l + 1][1] = shifts[l * 2 + 1];
  }
  const float wf0 = Wf[0], wf1 = Wf[1], bfv = bf[0];
  const float NEG_LOG2E = -1.4426950408889634f;  // GAMMA == 1.0

  const int lane = threadIdx.x;      // 0..31, one wave per workgroup
  const int stride = gridDim.x;
  const unsigned totalFloats = (unsigned)N * 2u;

  int t = blockIdx.x;
  int buf = 0;

#if USE_TDM
  if (t < numTiles) {
    unsigned fstart = (unsigned)t * TILE_FLOATS;
    tdm_issue_load(0u, x + fstart, TILE_FLOATS, totalFloats - fstart);
    tdm_issue_load(TILE_BYTES, y + fstart, TILE_FLOATS, totalFloats - fstart);
  }
#endif

  for (; t < numTiles; t += stride) {
#if USE_TDM
    int nextT = t + stride;
    if (nextT < numTiles) {
      unsigned nb = (unsigned)(buf ^ 1) * BUF_STRIDE_BYTES;
      unsigned fstart = (unsigned)nextT * TILE_FLOATS;
      tdm_issue_load(nb, x + fstart, TILE_FLOATS, totalFloats - fstart);
      tdm_issue_load(nb + TILE_BYTES, y + fstart, TILE_FLOATS,
                     totalFloats - fstart);
      wait_tensor<2>();  // current buffer's 2 DMAs complete (in-order)
    } else {
      wait_tensor<0>();
    }
#endif
    const int tileStart = t * TILE_PAIRS;
#pragma unroll
    for (int s = 0; s < TILE_PAIRS / 64; ++s) {  // 64 pairs per wave step
      const int idx = s * 32 + lane;             // 16B-granule index in tile
#if USE_TDM
      const int fi = buf * BUF_STRIDE_FLOATS + idx * 4;
      v4f xv = *(const v4f*)&smbase[fi];                 // ds_load_b128
      v4f yv = *(const v4f*)&smbase[fi + TILE_FLOATS];   // ds_load_b128
#else
      const size_t gfi = (size_t)tileStart * 2 + (size_t)idx * 4;
      v4f xv = __builtin_nontemporal_load((const v4f*)(x + gfi));
      v4f yv = __builtin_nontemporal_load((const v4f*)(y + gfi));
#endif
      const int p0 = tileStart + idx * 2;  // first of 2 pairs this lane owns

      // pair a
      float d0 = xv.x - yv.x, d1 = xv.y - yv.y;
      float r2a = fmaf(d1, d1, d0 * d0);
      float rbfa = fast_exp2(NEG_LOG2E * r2a);
      float fxa = fraud_net(xv.x, xv.y, Wp, Bp, Sp, Tp, wf0, wf1, bfv);
      float fya = fraud_net(yv.x, yv.y, Wp, Bp, Sp, Tp, wf0, wf1, bfv);
      float ra = rbfa * fxa * fya;

      // pair b
      float e0 = xv.z - yv.z, e1 = xv.w - yv.w;
      float r2b = fmaf(e1, e1, e0 * e0);
      float rbfb = fast_exp2(NEG_LOG2E * r2b);
      float fxb = fraud_net(xv.z, xv.w, Wp, Bp, Sp, Tp, wf0, wf1, bfv);
      float fyb = fraud_net(yv.z, yv.w, Wp, Bp, Sp, Tp, wf0, wf1, bfv);
      float rb = rbfb * fxb * fyb;

      if (p0 + 1 < N) {
        v2f r; r.x = ra; r.y = rb;
        __builtin_nontemporal_store(r, (v2f*)(out + p0));
      } else if (p0 < N) {
        out[p0] = ra;
      }
    }
    buf ^= 1;
  }
}

extern "C" void kernel_launch(void* const* d_in, const int* in_sizes, int n_in,
                              void* d_out, int out_size, void* d_ws,
                              size_t ws_size, hipStream_t stream) {
  const float* x = (const float*)d_in[0];
  const float* y = (const float*)d_in[1];
  const float* W0 = (const float*)d_in[2];
  const float* b0 = (const float*)d_in[3];
  const float* scale0 = (const float*)d_in[4];
  const float* shift0 = (const float*)d_in[5];
  const float* Ws = (const float*)d_in[6];
  const float* bs = (const float*)d_in[7];
  const float* scales = (const float*)d_in[8];
  const float* shifts = (const float*)d_in[9];
  const float* Wf = (const float*)d_in[10];
  const float* bf = (const float*)d_in[11];
  float* out = (float*)d_out;

  const int N = in_sizes[0] / 2;  // x is [N,2]
  if (N <= 0) return;
  const int numTiles = (N + TILE_PAIRS - 1) / TILE_PAIRS;
  int blocks = numTiles < 4096 ? numTiles : 4096;  // ~4k waves in flight

  // NOTE: pass dynamic-LDS size unconditionally. The host pass cannot see
  // device-side __has_builtin results, so USE_TDM may differ between the two
  // compilation passes; 16KB of dynamic LDS is harmless for the fallback.
  fraud_rbf_kernel<<<blocks, 32, SHMEM_BYTES, stream>>>(
      x, y, W0, b0, scale0, shift0, Ws, bs, scales, shifts, Wf, bf, out, N,
      numTiles);
}